// Eq1to3_26817775796965
// MI455X (gfx1250) — compile-verified
//
#include <hip/hip_runtime.h>

// N=16 batch, D=64 in_dim, S=32 out_dim, M=48 set size, BASIS=4
// out[n,s,i,j,k] = t0[n,s,i] + t1[n,s,j] + t2[n,s,k] + (i==j==k)*t3[n,s,i] + bias[s]
// where t[b,n,s,m] = sum_d coefs[d,s,b] * x[n,d,m]

typedef float v2f __attribute__((ext_vector_type(2)));
typedef float v8f __attribute__((ext_vector_type(8)));
typedef float f4  __attribute__((ext_vector_type(4)));

#define NB   16
#define DD   64
#define SS   32
#define MM   48
#define RR   128   // S*BASIS rows, r = s*4 + b

// ---------------------------------------------------------------------------
// Phase 1: t[n][r][m] = sum_d A[r][d] * x[n][d][m],  A[r][d] = coefs[d*128 + r]
// One wave (32 lanes, EXEC all ones) per (n, col-tile). 8 row-tiles x 16 K-steps
// of V_WMMA_F32_16X16X4_F32 each.
//
// fp32 WMMA fragment layouts (ISA 7.12.2):
//   A 16x4:  VGPR0 = K0 (lanes 0-15) / K2 (lanes 16-31); VGPR1 = K1 / K3; M = lane%16
//   B 4x16:  VGPR0 = K0 row / K2 row; VGPR1 = K1 / K3; N = lane%16
//   C 16x16: VGPR v = row v (lanes 0-15) / row v+8 (lanes 16-31); N = lane%16
// ---------------------------------------------------------------------------
__global__ void eq1to3_phase1_wmma(const float* __restrict__ x,      // (16,64,48)
                                   const float* __restrict__ coefs,  // (64,32,4)
                                   float* __restrict__ t)            // (16,128,48)
{
    const int n    = blockIdx.x / 3;       // batch
    const int ct   = blockIdx.x % 3;       // column tile over m (3 x 16 = 48)
    const int lane = threadIdx.x;          // 0..31
    const int hs   = lane >> 4;            // half select
    const int l16  = lane & 15;

    const float* xb = x + n * (DD * MM);
    float*       tn = t + n * (RR * MM);
    const int colBase = ct * 16;

    for (int rt = 0; rt < 8; ++rt) {       // 8 row tiles cover R=128
        const int rowBase = rt * 16;
        v8f acc = {};
        #pragma unroll
        for (int kk = 0; kk < 16; ++kk) {  // K = 64 in steps of 4
            const int d0 = kk * 4 + hs * 2;
            v2f a, b;
            // A: row (rowBase + l16), K = d0, d0+1
            a[0] = coefs[(d0 + 0) * RR + rowBase + l16];
            a[1] = coefs[(d0 + 1) * RR + rowBase + l16];
            // B: col (colBase + l16), K rows d0, d0+1
            b[0] = xb[(d0 + 0) * MM + colBase + l16];
            b[1] = xb[(d0 + 1) * MM + colBase + l16];
            acc = __builtin_amdgcn_wmma_f32_16x16x4_f32(
                      /*neg_a=*/false, a, /*neg_b=*/false, b,
                      /*c_mod=*/(short)0, acc,
                      /*reuse_a=*/false, /*reuse_b=*/false);
        }
        #pragma unroll
        for (int v = 0; v < 8; ++v) {
            const int row = rowBase + v + hs * 8;
            tn[row * MM + colBase + l16] = acc[v];
        }
    }
}

// ---------------------------------------------------------------------------
// Phase 2: streaming broadcast expansion. 226.5 MB of stores -> HBM-bound.
// One 192-thread block per (n, s, i) plane: 48*48 floats = 576 float4,
// exactly 3 non-temporal 128-bit stores per thread (NT: output > L2, never
// re-read, keep it out of the 192MB L2).
// ---------------------------------------------------------------------------
__global__ void eq1to3_phase2(const float* __restrict__ t,     // (16,128,48), r=s*4+b
                              const float* __restrict__ bias,  // (32)
                              float* __restrict__ out)         // (16,32,48,48,48)
{
    const int blk = blockIdx.x;
    const int i   = blk % MM;
    const int ns  = blk / MM;
    const int s   = ns % SS;
    const int n   = ns / SS;

    const float* tb = t + ((size_t)n * RR + (size_t)s * 4) * MM; // tb[b*48 + m]

    __shared__ float s_t1[MM];
    __shared__ float s_t2[MM];
    __shared__ float s_sc[3];   // t0[i]+bias precursors: t0[i], t3[i], bias[s]

    const int tid = threadIdx.x;
    if (tid < MM)            s_t1[tid]      = tb[1 * MM + tid];
    else if (tid < 2 * MM)   s_t2[tid - MM] = tb[2 * MM + (tid - MM)];
    else if (tid == 2 * MM)     s_sc[0] = tb[i];            // t0[i]
    else if (tid == 2 * MM + 1) s_sc[1] = tb[3 * MM + i];   // t3[i]
    else if (tid == 2 * MM + 2) s_sc[2] = bias[s];
    __syncthreads();

    const float base_i = s_sc[0] + s_sc[2];   // t0[i] + bias
    const float t3i    = s_sc[1];

    f4* __restrict__ outp =
        (f4*)(out + ((((size_t)n * SS + s) * MM + i) * MM) * MM);

    const int iQuad = i >> 2;
    const int iLane = i & 3;

    // 576 = 192 threads * 3 iterations, no tail.
    #pragma unroll
    for (int it = 0; it < 3; ++it) {
        const int q  = tid + it * 192;
        const int j  = q / 12;        // row within plane
        const int k4 = q % 12;        // float4 index along k
        const float tj = base_i + s_t1[j];
        const int k0 = k4 * 4;
        f4 val;
        val[0] = tj + s_t2[k0 + 0];
        val[1] = tj + s_t2[k0 + 1];
        val[2] = tj + s_t2[k0 + 2];
        val[3] = tj + s_t2[k0 + 3];
        if (j == i && k4 == iQuad) {
            val[iLane] += t3i;        // superdiagonal i==j==k
        }
        __builtin_nontemporal_store(val, &outp[q]);
    }
}

extern "C" void kernel_launch(void* const* d_in, const int* in_sizes, int n_in,
                              void* d_out, int out_size, void* d_ws, size_t ws_size,
                              hipStream_t stream) {
    const float* x     = (const float*)d_in[0];   // (16,64,48)
    const float* coefs = (const float*)d_in[1];   // (64,32,4)
    const float* bias  = (const float*)d_in[2];   // (1,32,1,1,1)
    float* out = (float*)d_out;                   // (16,32,48,48,48)
    float* t   = (float*)d_ws;                    // 16*128*48 floats = 384 KB

    // Phase 1: tiny WMMA GEMM producing t[b,n,s,m]
    eq1to3_phase1_wmma<<<NB * 3, 32, 0, stream>>>(x, coefs, t);

    // Phase 2: HBM-bound broadcast (one block per (n,s,i) plane)
    eq1to3_phase2<<<NB * SS * MM, 192, 0, stream>>>(t, bias, out);
}